// TCNGCNUnit_74990128988683
// MI455X (gfx1250) — compile-verified
//
#include <hip/hip_runtime.h>
#include <hip/hip_bf16.h>
#include <math.h>
#include <stdint.h>

typedef __attribute__((ext_vector_type(2))) float v2f;
typedef __attribute__((ext_vector_type(8))) float v8f;

#define VV   25
#define VT   29
#define TOPKN 9
#define EPSF 1e-8f
#define BNE  1e-5f

// ---------------- workspace layout (float offsets) ----------------
#define WS_AALL 0                          // 256 * 32 * 32
#define WS_ASUM (WS_AALL + 256*32*32)      // 256 * 32
#define WS_H1   (WS_ASUM + 256*32)         // 32 * 16 * 32
#define WS_INC  (WS_H1 + 32*16*32)         // 256 * 32 * 32
#define WS_WF   (WS_INC + 256*32*32)       // 64 * 64
#define WS_BF   (WS_WF + 64*64)            // 64
#define WS_Y    (WS_BF + 64)               // 32*64*512*25
#define WS_B    (WS_Y + 32*64*512*25)      // 32*64*512*25

__device__ __forceinline__ v8f wmma4(v2f a, v2f b, v8f c) {
    return __builtin_amdgcn_wmma_f32_16x16x4_f32(false, a, false, b, (short)0, c,
                                                 false, false);
}

// async global->LDS copy helpers (CDNA5 GLOBAL_LOAD_ASYNC_TO_LDS_*, ASYNCcnt)
__device__ __forceinline__ void async_copy_b32(void* lds, const void* gptr) {
    unsigned l = (unsigned)(uintptr_t)lds;   // low 32 bits of flat LDS ptr = LDS offset
    asm volatile("global_load_async_to_lds_b32 %0, %1, off"
                 :: "v"(l), "v"(gptr) : "memory");
}
__device__ __forceinline__ void async_copy_b128(void* lds, const void* gptr) {
    unsigned l = (unsigned)(uintptr_t)lds;
    asm volatile("global_load_async_to_lds_b128 %0, %1, off"
                 :: "v"(l), "v"(gptr) : "memory");
}
__device__ __forceinline__ void async_wait0() {
    asm volatile("s_wait_asynccnt 0x0" ::: "memory");
}

// ---------------- fold branch 1x1 conv + BN into one 64x64 matrix ----------------
__global__ __launch_bounds__(256) void k_fold(
    const float* w1, const float* b1, const float* g1, const float* bb1,
    const float* w2, const float* b2, const float* g2, const float* bb2,
    const float* w3, const float* b3, const float* g3, const float* bb3,
    const float* w4, const float* b4, const float* g4, const float* bb4,
    float* Wf, float* bf) {
    const float rs = rsqrtf(1.0f + BNE);
    for (int idx = threadIdx.x; idx < 64 * 64; idx += 256) {
        int o = idx >> 6, c = idx & 63;
        int g = o >> 4, j = o & 15;
        const float* w  = (g == 0) ? w1 : (g == 1) ? w2 : (g == 2) ? w3 : w4;
        const float* gg = (g == 0) ? g1 : (g == 1) ? g2 : (g == 2) ? g3 : g4;
        Wf[o * 64 + c] = gg[j] * rs * w[j * 64 + c];
    }
    if (threadIdx.x < 64) {
        int o = threadIdx.x, g = o >> 4, j = o & 15;
        const float* b  = (g == 0) ? b1  : (g == 1) ? b2  : (g == 2) ? b3  : b4;
        const float* gg = (g == 0) ? g1  : (g == 1) ? g2  : (g == 2) ? g3  : g4;
        const float* bb = (g == 0) ? bb1 : (g == 1) ? bb2 : (g == 2) ? bb3 : bb4;
        bf[o] = gg[j] * rs * b[j] + bb[j];
    }
}

// ---------------- hypergraph stage 1: pooled mean, vproj, h1, top-9 incidence ----------------
__global__ __launch_bounds__(256) void k_hyper1(
    const float* x, const float* hj,
    const float* tv_w, const float* tv_b,
    const float* tw_w1, const float* tw_b1,
    float* h1buf, float* incbuf) {
    int nk = blockIdx.x, n = nk >> 3, k = nk & 7;
    int tid = threadIdx.x;
    __shared__ float pg[8][32];
    __shared__ float vp[32][2];

    {   // pooled mean over t (xcat = [x | broadcast hyper_joint])
        int i = tid >> 5, v = tid & 31;
        float s = 0.f;
        if (v < VV) {
            const float* px = x + ((size_t)(n * 64 + k * 8 + i) * 512) * 25 + v;
            for (int t = 0; t < 512; ++t) s += px[t * 25];
            s *= (1.0f / 512.0f);
        } else if (v < VT) {
            s = hj[(v - VV) * 64 + k * 8 + i];
        }
        pg[i][v] = s;
    }
    __syncthreads();
    if (tid < VT * 2) {   // vproj and h1 (HID = 2)
        int vv = tid >> 1, h = tid & 1;
        float a = tv_b[k * 2 + h];
        float b = tw_b1[k * 2 + h];
        for (int ii = 0; ii < 8; ++ii) {
            float p = pg[ii][vv];
            a += p * tv_w[(k * 2 + h) * 8 + ii];
            b += p * tw_w1[(k * 2 + h) * 8 + ii];
        }
        vp[vv][h] = a;
        h1buf[(n * 16 + (k * 2 + h)) * 32 + vv] = (b >= 0.f) ? b : 0.01f * b;
    }
    __syncthreads();
    if (tid < 32) {   // per-row top-9 of -dist + softmax -> incidence
        int u = tid;
        float* row = incbuf + ((size_t)nk * 32 + u) * 32;
        for (int vv = 0; vv < 32; ++vv) row[vv] = 0.f;
        if (u < VT) {
            float negv[VT];
            float u0 = vp[u][0], u1 = vp[u][1];
            for (int vv = 0; vv < VT; ++vv) {
                float d0 = u0 - vp[vv][0], d1 = u1 - vp[vv][1];
                negv[vv] = -sqrtf(fmaxf(d0 * d0 + d1 * d1, 1e-12f));
            }
            unsigned mask = 0; float vals[TOPKN]; int sel[TOPKN];
            for (int j = 0; j < TOPKN; ++j) {
                float best = -1e30f; int bi = 0;
                for (int vv = 0; vv < VT; ++vv)
                    if (!((mask >> vv) & 1u) && negv[vv] > best) { best = negv[vv]; bi = vv; }
                mask |= 1u << bi; vals[j] = best; sel[j] = bi;
            }
            float m = vals[0], den = 0.f;
            for (int j = 0; j < TOPKN; ++j) den += expf(vals[j] - m);
            float inv = 1.0f / den;
            for (int j = 0; j < TOPKN; ++j) row[sel[j]] = expf(vals[j] - m) * inv;
        }
    }
}

// ---------------- hypergraph stage 2: w, col, Hw, hyper_adj, Aall ----------------
__global__ __launch_bounds__(32) void k_hyper2(
    const float* adj, const float* ei, const float* alpha,
    const float* tw_w2, const float* tw_b2,
    const float* h1buf, const float* incbuf,
    float* Aall, float* asum) {
    int nk = blockIdx.x, n = nk >> 3, k = nk & 7;
    int tid = threadIdx.x;
    __shared__ float inc_s[VT][32];
    __shared__ float w_s[32];
    __shared__ float wcol[32];
    const float* incb = incbuf + (size_t)nk * 32 * 32;
    for (int idx = tid; idx < VT * 32; idx += 32) inc_s[idx >> 5][idx & 31] = incb[idx];
    {
        float wv = 0.f;
        if (tid < VT) {
            wv = tw_b2[k];
            for (int c = 0; c < 16; ++c)
                wv += tw_w2[k * 16 + c] * h1buf[(n * 16 + c) * 32 + tid];
            wv = tanhf(wv);
        }
        w_s[tid] = wv;
    }
    __syncthreads();
    {
        float col = EPSF;
        if (tid < VT) for (int u = 0; u < VT; ++u) col += fabsf(inc_s[u][tid]);
        wcol[tid] = (tid < VT) ? w_s[tid] / col : 0.f;
    }
    __syncthreads();
    float ra = fmaxf(alpha[0], 0.f);
    int u = tid;
    float* arow = Aall + ((size_t)nk * 32 + u) * 32;
    if (u < VT) {
        float Hwr[VT];
        float nrm = EPSF;
        for (int e = 0; e < VT; ++e) { float hw = inc_s[u][e] * w_s[e]; Hwr[e] = hw; nrm += fabsf(hw); }
        float invn = 1.0f / nrm;
        for (int e = 0; e < VT; ++e) Hwr[e] *= invn * wcol[e];
        const float* Ar = adj + (size_t)k * VT * VT + (size_t)u * VT;
        const float* Er = ei  + (size_t)k * VT * VT + (size_t)u * VT;
        float rsum = EPSF;
        for (int vv = 0; vv < VT; ++vv) rsum += fabsf(Ar[vv] * Er[vv]);
        float rinv = 1.0f / rsum;
        float su = 0.f;
        for (int vv = 0; vv < 32; ++vv) {
            float val = 0.f;
            if (vv < VT) {
                float ha = 0.f;
                for (int e = 0; e < VT; ++e) ha += Hwr[e] * inc_s[vv][e];
                val = Ar[vv] * Er[vv] * rinv + ra * ha;
            }
            arow[vv] = val; su += val;
        }
        asum[nk * 32 + u] = su;
    } else {
        for (int vv = 0; vv < 32; ++vv) arow[vv] = 0.f;
        asum[nk * 32 + u] = 0.f;
    }
}

// ---------------- GCN core: dense = cd_w @ xcat ; y = dense @ Aall^T ; BN+res+relu ----------------
__global__ __launch_bounds__(256) void k_gcn(
    const float* x, const float* hj, const float* cd_w, const float* cd_b,
    const float* bn_g, const float* bn_b,
    const float* Aall, const float* asum, float* yout) {
    int bx = blockIdx.x;
    int nk = bx >> 5, tc = bx & 31;
    int n = nk >> 3, k = nk & 7;
    int t0 = tc * 16;
    int tid = threadIdx.x;
    int wv = tid >> 5, lane = tid & 31;
    int half = lane >> 4, lm = lane & 15;
    __shared__ float Xs[8][16][32];
    __shared__ float Ds[8][16][34];   // padded for bank-conflict-free strided DS loads
    __shared__ float As[32][34];
    __shared__ float asums[32];
    __shared__ float cw[64];
    // real joints: async global->LDS (ASYNCcnt path), 3200 B32 transfers
    for (int idx = tid; idx < 8 * 16 * 25; idx += 256) {
        int i = idx / 400, r = idx % 400, t = r / 25, vv = r % 25;
        const float* g = x + ((size_t)(n * 64 + k * 8 + i) * 512 + t0 + t) * 25 + vv;
        async_copy_b32(&Xs[i][t][vv], g);
    }
    {   // virtual joints + zero padding via normal DS stores
        int i = tid >> 5, vv = tid & 31;
        if (vv >= VV) {
            float hval = (vv < VT) ? hj[(vv - VV) * 64 + k * 8 + i] : 0.f;
            for (int t = 0; t < 16; ++t) Xs[i][t][vv] = hval;
        }
    }
    for (int idx = tid; idx < 1024; idx += 256)
        As[idx >> 5][idx & 31] = Aall[(size_t)nk * 1024 + idx];
    if (tid < 32) asums[tid] = asum[nk * 32 + tid];
    if (tid < 64) cw[tid] = cd_w[k * 64 + tid];
    async_wait0();
    __syncthreads();
    {   // dense: 8x8 channel mix (small -> VALU)
        int c = tid >> 5, vv = tid & 31;
        for (int t = 0; t < 16; ++t) {
            float a = 0.f;
            for (int i = 0; i < 8; ++i) a += cw[c * 8 + i] * Xs[i][t][vv];
            Ds[c][t][vv] = a;
        }
    }
    __syncthreads();
    // WMMA: rows = t (16 per wave, channel c = wave), K = v (32), N = u (2 tiles)
    int c = wv;
    v8f acc0 = {}, acc1 = {};
    for (int ks = 0; ks < 8; ++ks) {
        int kk = 4 * ks + 2 * half;                 // 16x4 layout: VGPR0=K{0,2}, VGPR1=K{1,3}
        v2f a  = *(const v2f*)&Ds[c][lm][kk];       // A[M=lm][K]
        v2f b0 = *(const v2f*)&As[lm][kk];          // B[K][N=lm]      = Aall[u=lm][v=K]
        v2f b1 = *(const v2f*)&As[lm + 16][kk];     // B[K][N=lm+16]
        acc0 = wmma4(a, b0, acc0);
        acc1 = wmma4(a, b1, acc1);
    }
    const float rsb = rsqrtf(1.0f + BNE);
    int ch = k * 8 + c;
    float sc = bn_g[ch] * rsb, bb = bn_b[ch], cb = cd_b[ch];
    const float* xr = x    + (size_t)(n * 64 + ch) * 512 * 25;
    float*       yr = yout + (size_t)(n * 64 + ch) * 512 * 25;
    for (int j = 0; j < 8; ++j) {
        int t = t0 + j + 8 * half;                  // C layout: VGPR j -> M=j / j+8
        {
            int u = lm;
            float val = acc0[j] + cb * asums[u];
            yr[t * 25 + u] = fmaxf(val * sc + bb + xr[t * 25 + u], 0.f);
        }
        int u = 16 + lm;
        if (u < VV) {
            float val = acc1[j] + cb * asums[u];
            yr[t * 25 + u] = fmaxf(val * sc + bb + xr[t * 25 + u], 0.f);
        }
    }
}

// ---------------- fused 4-branch 1x1 conv: 128x64x64 fp32 WMMA GEMM per tile ----------------
__global__ __launch_bounds__(256) void k_branch(
    const float* y, const float* Wf, const float* bf, float* bout) {
    int bx = blockIdx.x;
    int n = bx / 100, mc = bx % 100;
    int m0 = mc * 128;
    int tid = threadIdx.x, wv = tid >> 5, lane = tid & 31;
    int half = lane >> 4, lm = lane & 15;
    __shared__ float Ys[64][132];   // [c][m]; row = 512B = one wave of b128 async loads
    __shared__ float Ws[64][66];    // [o][c]
    __shared__ float bfs[64];
    // y tile: async global->LDS b128 (16B aligned on both sides)
    for (int c = wv; c < 64; c += 8)
        async_copy_b128(&Ys[c][lane * 4],
                        y + (size_t)(n * 64 + c) * 12800 + m0 + lane * 4);
    for (int idx = tid; idx < 4096; idx += 256) Ws[idx >> 6][idx & 63] = Wf[idx];
    if (tid < 64) bfs[tid] = bf[tid];
    async_wait0();
    __syncthreads();
    v8f acc[4] = {};
    int mrow = 16 * wv + lm;
    for (int ks = 0; ks < 16; ++ks) {
        int kk = 4 * ks + 2 * half;
        v2f a;                                      // A[M=m][K=c] from [c][m] layout
        a.x = Ys[kk][mrow];
        a.y = Ys[kk + 1][mrow];
#pragma unroll
        for (int q = 0; q < 4; ++q) {
            v2f b = *(const v2f*)&Ws[q * 16 + lm][kk];   // B[K=c][N=o] = Wf[o][c]
            acc[q] = wmma4(a, b, acc[q]);
        }
    }
#pragma unroll
    for (int q = 0; q < 4; ++q) {
        int o = q * 16 + lm;
        float bias = bfs[o];
        float* br = bout + (size_t)(n * 64 + o) * 12800 + m0;
        for (int j = 0; j < 8; ++j) {
            int m = 16 * wv + j + 8 * half;
            float val = acc[q][j] + bias;
            if (o < 48) val = fmaxf(val, 0.f);      // branches 1-3 have ReLU; branch 4 does not
            br[m] = val;
        }
    }
}

// ---------------- temporal convs / maxpool / residual / final relu ----------------
__global__ __launch_bounds__(256) void k_temporal(
    const float* bpre, const float* x,
    const float* tc1_w, const float* tc1_b, const float* tc1_g, const float* tc1_bb,
    const float* tc2_w, const float* tc2_b, const float* tc2_g, const float* tc2_bb,
    const float* p3_g, const float* p3_b,
    float* out) {
    int bx = blockIdx.x;
    int no = bx / 50, chunk = bx % 50;
    int n = no >> 6, o = no & 63;
    int m = chunk * 256 + threadIdx.x;
    int t = m / 25, v = m % 25;
    int g = o >> 4, j = o & 15;
    const float rsb = rsqrtf(1.0f + BNE);
    float val;
    if (g == 0) {           // tconv k=5 dil=1, edge padding
        float s = tc1_b[j];
        const float* base = bpre + (size_t)n * 64 * 12800;
        for (int c = 0; c < 16; ++c) {
            const float* pc = base + (size_t)c * 12800;
            for (int kt = 0; kt < 5; ++kt) {
                int tt = min(max(t - 2 + kt, 0), 511);
                s += tc1_w[(j * 16 + c) * 5 + kt] * pc[tt * 25 + v];
            }
        }
        val = s * tc1_g[j] * rsb + tc1_bb[j];
    } else if (g == 1) {    // tconv k=5 dil=2, edge padding
        float s = tc2_b[j];
        const float* base = bpre + (size_t)(n * 64 + 16) * 12800;
        for (int c = 0; c < 16; ++c) {
            const float* pc = base + (size_t)c * 12800;
            for (int kt = 0; kt < 5; ++kt) {
                int tt = min(max(t - 4 + 2 * kt, 0), 511);
                s += tc2_w[(j * 16 + c) * 5 + kt] * pc[tt * 25 + v];
            }
        }
        val = s * tc2_g[j] * rsb + tc2_bb[j];
    } else if (g == 2) {    // maxpool3 over t (window clipped at edges) + second BN
        const float* pc = bpre + (size_t)(n * 64 + 32 + j) * 12800;
        float s = pc[t * 25 + v];
        if (t > 0)   s = fmaxf(s, pc[(t - 1) * 25 + v]);
        if (t < 511) s = fmaxf(s, pc[(t + 1) * 25 + v]);
        val = s * p3_g[j] * rsb + p3_b[j];
    } else {                // branch 4 passthrough (BN already applied)
        val = bpre[(size_t)(n * 64 + 48 + j) * 12800 + m];
    }
    out[(size_t)no * 12800 + m] = fmaxf(val + x[(size_t)no * 12800 + m], 0.f);
}

__global__ void k_copyhj(const float* hj, float* out) {
    if (threadIdx.x < 256) out[26214400 + threadIdx.x] = hj[threadIdx.x];
}

extern "C" void kernel_launch(void* const* d_in, const int* in_sizes, int n_in,
                              void* d_out, int out_size, void* d_ws, size_t ws_size,
                              hipStream_t stream) {
    const float* x      = (const float*)d_in[0];
    const float* hj     = (const float*)d_in[1];
    const float* alpha  = (const float*)d_in[2];
    const float* adj    = (const float*)d_in[3];
    const float* ei     = (const float*)d_in[4];
    const float* tv_w   = (const float*)d_in[5];
    const float* tv_b   = (const float*)d_in[6];
    const float* tw_w1  = (const float*)d_in[7];
    const float* tw_b1  = (const float*)d_in[8];
    const float* tw_w2  = (const float*)d_in[9];
    const float* tw_b2  = (const float*)d_in[10];
    const float* cd_w   = (const float*)d_in[11];
    const float* cd_b   = (const float*)d_in[12];
    const float* bn_g   = (const float*)d_in[13];
    const float* bn_b   = (const float*)d_in[14];
    const float* br1_w  = (const float*)d_in[15];
    const float* br1_b  = (const float*)d_in[16];
    const float* br1_g  = (const float*)d_in[17];
    const float* br1_bb = (const float*)d_in[18];
    const float* tc1_w  = (const float*)d_in[19];
    const float* tc1_b  = (const float*)d_in[20];
    const float* tc1_g  = (const float*)d_in[21];
    const float* tc1_bb = (const float*)d_in[22];
    const float* br2_w  = (const float*)d_in[23];
    const float* br2_b  = (const float*)d_in[24];
    const float* br2_g  = (const float*)d_in[25];
    const float* br2_bb = (const float*)d_in[26];
    const float* tc2_w  = (const float*)d_in[27];
    const float* tc2_b  = (const float*)d_in[28];
    const float* tc2_g  = (const float*)d_in[29];
    const float* tc2_bb = (const float*)d_in[30];
    const float* br3_w  = (const float*)d_in[31];
    const float* br3_b  = (const float*)d_in[32];
    const float* br3_g  = (const float*)d_in[33];
    const float* br3_bb = (const float*)d_in[34];
    const float* p3_g   = (const float*)d_in[35];
    const float* p3_b   = (const float*)d_in[36];
    const float* br4_w  = (const float*)d_in[37];
    const float* br4_b  = (const float*)d_in[38];
    const float* br4_g  = (const float*)d_in[39];
    const float* br4_bb = (const float*)d_in[40];

    float* W = (float*)d_ws;
    float* out = (float*)d_out;

    k_fold<<<1, 256, 0, stream>>>(br1_w, br1_b, br1_g, br1_bb,
                                  br2_w, br2_b, br2_g, br2_bb,
                                  br3_w, br3_b, br3_g, br3_bb,
                                  br4_w, br4_b, br4_g, br4_bb,
                                  W + WS_WF, W + WS_BF);
    k_hyper1<<<256, 256, 0, stream>>>(x, hj, tv_w, tv_b, tw_w1, tw_b1,
                                      W + WS_H1, W + WS_INC);
    k_hyper2<<<256, 32, 0, stream>>>(adj, ei, alpha, tw_w2, tw_b2,
                                     W + WS_H1, W + WS_INC,
                                     W + WS_AALL, W + WS_ASUM);
    k_gcn<<<256 * 32, 256, 0, stream>>>(x, hj, cd_w, cd_b, bn_g, bn_b,
                                        W + WS_AALL, W + WS_ASUM, W + WS_Y);
    k_branch<<<32 * 100, 256, 0, stream>>>(W + WS_Y, W + WS_WF, W + WS_BF, W + WS_B);
    k_temporal<<<32 * 64 * 50, 256, 0, stream>>>(W + WS_B, x,
                                                 tc1_w, tc1_b, tc1_g, tc1_bb,
                                                 tc2_w, tc2_b, tc2_g, tc2_bb,
                                                 p3_g, p3_b, out);
    k_copyhj<<<1, 256, 0, stream>>>(hj, out);
}